// CharTiedEncoderDecoder_46918222742060
// MI455X (gfx1250) — compile-verified
//
#include <hip/hip_runtime.h>

// CDNA5 / gfx1250, wave32. Char-CNN word encoder:
//   per word: gather 20 chars -> 20x32 f32 emb (+1 const channel) -> conv1d(512, k=4, valid)
//   -> maxpool over 17 positions. Cast as GEMM Y[17x512] = X[17x132] * W^T via
//   V_WMMA_F32_16X16X4_F32 (f32 exact path; problem is HBM-bound so no need for bf16).

typedef __attribute__((ext_vector_type(2))) float v2f;
typedef __attribute__((ext_vector_type(8))) float v8f;

#define NCH   20          // chars per word
#define CE    32          // char embedding dim
#define KC    33          // channels incl. lang bit
#define WEMB  512         // output (conv) channels
#define KDIM  132         // im2col K = KC*4
#define NTILES (WEMB/16)  // 32

// LDS weight layout: j-pairs. element (j, we) at (j>>1)*PAIR_STRIDE + 2*we + (j&1).
// PAIR_STRIDE=1056 floats (>=1024, and 1056%64==32 -> half-waves hit disjoint banks).
#define PAIR_STRIDE 1056
#define LDS_FLOATS  (66 * PAIR_STRIDE)   // 66 pair-rows * 1056 = 69696 floats = 278784 B

__global__ __launch_bounds__(256) void char_cnn_wmma(
    const float* __restrict__ char_emb_w,   // [256, 32]
    const float* __restrict__ cnn_w,        // [512, 33, 4]  flat: we*132 + (c*4 + k)
    const int*   __restrict__ data,         // [nWords]
    const int*   __restrict__ spelling,     // [V, 20]
    float*       __restrict__ out,          // [nWords, 512]
    int nWords)
{
  extern __shared__ float ldsW[];
  const int tid = threadIdx.x;

  // Stage weights transposed+pair-interleaved into LDS. Global read coalesced.
  for (int e = tid; e < WEMB * KDIM; e += 256) {
    const int we = e / KDIM;
    const int j  = e % KDIM;
    ldsW[(j >> 1) * PAIR_STRIDE + 2 * we + (j & 1)] = cnn_w[e];
  }
  __syncthreads();

  const int lane    = tid & 31;
  const int wave    = tid >> 5;
  const int m_local = lane & 15;   // M row within tile / N col within tile
  const int half    = lane >> 4;   // selects K sub-pair

  for (int word = blockIdx.x * 8 + wave; word < nWords; word += gridDim.x * 8) {
    const int wid   = data[word];
    const int sbase = wid * NCH;

    // A fragments: a[mt][c] for M-tiles mt=0 (t=0..15) and mt=1 (t=16..31, padded).
    // Layout of V_WMMA_F32_16X16X4_F32 A (16x4): lane<16 -> K0,K1 ; lane>=16 -> K2,K3.
    // k-tile c needs X[t, 4c+klocal] = E[t + klocal, c], klocal = 2*half + {0,1}.
    v2f a[2][KC];
    #pragma unroll
    for (int mt = 0; mt < 2; ++mt) {
      const int r1 = mt * 16 + m_local + 2 * half;   // embedding row for .x
      const int r2 = r1 + 1;                         // embedding row for .y
      const bool ok1 = (r1 < NCH);
      const bool ok2 = (r2 < NCH);
      const int ch1 = ok1 ? spelling[sbase + r1] : 0;
      const int ch2 = ok2 ? spelling[sbase + r2] : 0;
      const float* e1 = char_emb_w + ch1 * CE;
      const float* e2 = char_emb_w + ch2 * CE;
      #pragma unroll
      for (int cb = 0; cb < CE; cb += 4) {
        const float4 q1 = ok1 ? *(const float4*)(e1 + cb) : make_float4(0.f, 0.f, 0.f, 0.f);
        const float4 q2 = ok2 ? *(const float4*)(e2 + cb) : make_float4(0.f, 0.f, 0.f, 0.f);
        a[mt][cb + 0].x = q1.x;  a[mt][cb + 0].y = q2.x;
        a[mt][cb + 1].x = q1.y;  a[mt][cb + 1].y = q2.y;
        a[mt][cb + 2].x = q1.z;  a[mt][cb + 2].y = q2.z;
        a[mt][cb + 3].x = q1.w;  a[mt][cb + 3].y = q2.w;
      }
      a[mt][CE].x = ok1 ? 1.f : 0.f;   // lang-bit channel
      a[mt][CE].y = ok2 ? 1.f : 0.f;
    }

    // GEMM over 32 N-tiles of 16 output channels each; K = 33 steps of 4.
    #pragma unroll 1
    for (int nt = 0; nt < NTILES; ++nt) {
      v8f acc0 = {};   // t = 0..15
      v8f acc1 = {};   // t = 16..31 (only t=16 valid)
      const int colBase = 2 * (nt * 16 + m_local);
      #pragma unroll
      for (int kt = 0; kt < KC; ++kt) {
        const int pair = 2 * kt + half;
        const v2f b = *(const v2f*)&ldsW[pair * PAIR_STRIDE + colBase];
        acc0 = __builtin_amdgcn_wmma_f32_16x16x4_f32(
            false, a[0][kt], false, b, (short)0, acc0, false, false);
        acc1 = __builtin_amdgcn_wmma_f32_16x16x4_f32(
            false, a[1][kt], false, b, (short)0, acc1, false, false);
      }
      // Max-pool over t=0..16.
      // C/D layout: VGPR i, lane<16 -> (M=i, N=lane); lane>=16 -> (M=i+8, N=lane-16).
      float m = acc0[0];
      #pragma unroll
      for (int i = 1; i < 8; ++i) m = fmaxf(m, acc0[i]);
      m = fmaxf(m, __shfl_xor(m, 16, 32));            // merge t=0..7 with t=8..15
      const float t16 = __shfl(acc1[0], m_local, 32); // t=16 lives in lanes 0..15
      m = fmaxf(m, t16);
      if (lane < 16) out[word * WEMB + nt * 16 + lane] = m;
    }
  }
}

extern "C" void kernel_launch(void* const* d_in, const int* in_sizes, int n_in,
                              void* d_out, int out_size, void* d_ws, size_t ws_size,
                              hipStream_t stream) {
  const float* char_emb_w = (const float*)d_in[0];
  const float* cnn_w      = (const float*)d_in[1];
  const int*   data       = (const int*)d_in[2];
  const int*   spelling   = (const int*)d_in[3];
  float*       out        = (float*)d_out;
  const int nWords = in_sizes[2];        // B*S = 16384

  const int blocks = 256;                // grid-stride over words, 8 words per block pass
  const size_t shmem = LDS_FLOATS * sizeof(float);  // 278784 B < 320 KB/WGP
  hipLaunchKernelGGL(char_cnn_wmma, dim3(blocks), dim3(256), shmem, stream,
                     char_emb_w, cnn_w, data, spelling, out, nWords);
}